// MiSCFilter_mxt_34291018891850
// MI455X (gfx1250) — compile-verified
//
#include <hip/hip_runtime.h>
#include <hip/hip_bf16.h>
#include <stdint.h>

#define NN 2
#define CC 64
#define HH 256
#define WW 256
#define HWSZ (HH*WW)
#define GG 4
#define GC 16
#define HI 258
#define WI 258
#define LDS_STRIDE 68   // 64 + 4 pad floats; 272B row = 17*16B (keeps b128 LDS dests 16B aligned)

// ---------------------------------------------------------------------------
// Zero-fill workspace (gives us the zero pad border for free; interior is
// overwritten by the transpose kernel).
// ---------------------------------------------------------------------------
__global__ __launch_bounds__(256) void zero_f4(float4* __restrict__ p, int n4) {
    int i = blockIdx.x * blockDim.x + threadIdx.x;
    int stride = gridDim.x * blockDim.x;
    float4 z; z.x = 0.f; z.y = 0.f; z.z = 0.f; z.w = 0.f;
    for (; i < n4; i += stride) p[i] = z;
}

// ---------------------------------------------------------------------------
// Pad + NCHW->NHWC transpose via LDS, using CDNA5 async global<->LDS movers.
// Block = 256 threads, tile = one input row segment: (n, y, x0..x0+63) x 64 ch.
//   Phase 1: global_load_async_to_lds_b128  (coalesced 256B reads per channel)
//   Phase 2: global_store_async_from_lds_b32 (consecutive lanes -> consecutive
//            channels -> contiguous 128B/256B store bursts)
// ---------------------------------------------------------------------------
__global__ __launch_bounds__(256) void pad_transpose(const float* __restrict__ img1,
                                                     float* __restrict__ img_t) {
    __shared__ float lds[CC * LDS_STRIDE];
    const int b  = blockIdx.x;
    const int xt = b & 3;           // W/64 = 4 tiles per row
    const int y  = (b >> 2) & 255;
    const int n  = b >> 10;
    const int x0 = xt * 64;
    const int t  = threadIdx.x;

    // low 32 bits of a generic LDS pointer == wave-relative LDS byte offset
    const unsigned lbase = (unsigned)(uintptr_t)(void*)lds;

    const int xq = t & 15;          // 16B quad within the 64-pixel row
    const int c0 = t >> 4;          // 0..15
#pragma unroll
    for (int k = 0; k < 4; ++k) {
        const int c = k * 16 + c0;
        const float* gp = img1 + (((size_t)(n * CC + c) * HH + y) * WW + x0 + xq * 4);
        const unsigned loff = lbase + (unsigned)((c * LDS_STRIDE + xq * 4) * 4);
        const uint64_t ga = (uint64_t)(uintptr_t)gp;
        asm volatile("global_load_async_to_lds_b128 %0, %1, off"
                     :: "v"(loff), "v"(ga) : "memory");
    }
    asm volatile("s_wait_asynccnt 0" ::: "memory");
    __syncthreads();

    const int c  = t & 63;          // consecutive lanes -> consecutive channels
    const int xb = (t >> 6) * 16;
#pragma unroll
    for (int xx = 0; xx < 16; ++xx) {
        const int xl = xb + xx;
        const unsigned loff = lbase + (unsigned)((c * LDS_STRIDE + xl) * 4);
        float* dp = img_t + ((size_t)(n * HI + y + 1) * WI + (x0 + xl + 1)) * CC + c;
        const uint64_t ga = (uint64_t)(uintptr_t)dp;
        asm volatile("global_store_async_from_lds_b32 %0, %1, off"
                     :: "v"(ga), "v"(loff) : "memory");
    }
    asm volatile("s_wait_asynccnt 0" ::: "memory");
}

// ---------------------------------------------------------------------------
// Main deformable-sampling kernel on NHWC padded image.
// Block = 256 threads = 4 channel-quads x 64 consecutive pixels (one group).
// Each thread owns 4 contiguous channels -> every gather is one b128 load,
// and the 4 lanes of a pixel cover a contiguous 64B segment.
// ---------------------------------------------------------------------------
__global__ __launch_bounds__(256) void deform_main(const float* __restrict__ img_t,
                                                   const float* __restrict__ kv,
                                                   const float* __restrict__ kh,
                                                   const float* __restrict__ off,
                                                   const float* __restrict__ wgt,
                                                   float* __restrict__ out) {
    const int b = blockIdx.x;
    const int pixblock = b & 1023;          // HW/64 = 1024
    const int g = (b >> 10) & 3;
    const int n = b >> 12;
    const int t = threadIdx.x;
    const int cq = t & 3;
    const int pl = t >> 2;                  // 0..63
    const int lin = pixblock * 64 + pl;     // == ho*256 + wo
    const int wo = lin & 255;
    const int ho = lin >> 8;
    const int pixoff = lin;
    const int cbase = g * GC + cq * 4;

    // hoist the separable kernel taps (3+3 scalar loads, broadcast in-wave)
    float kvv[3], khh[3];
#pragma unroll
    for (int j = 0; j < 3; ++j) kvv[j] = kv[(n * 3 + j) * HWSZ + pixoff];
#pragma unroll
    for (int i = 0; i < 3; ++i) khh[i] = kh[(n * 12 + g * 3 + i) * HWSZ + pixoff];

    float ax = 0.f, ay = 0.f, az = 0.f, aw = 0.f;

#pragma unroll
    for (int p = 0; p < 9; ++p) {
        const int i = p / 3, j = p % 3;
        const int q = g * 9 + p;
        const float ox = off[(n * 72 + 2 * q)     * HWSZ + pixoff];
        const float oy = off[(n * 72 + 2 * q + 1) * HWSZ + pixoff];
        const float m  = wgt[(n * 36 + q) * HWSZ + pixoff] * khh[i] * kvv[j];

        // padded-image sample coords: gx = wo+1+(i-1)+ox, gy = ho+1+(j-1)+oy
        const float gx = (float)(wo + i) + ox;
        const float gy = (float)(ho + j) + oy;
        const float x0f = floorf(gx), y0f = floorf(gy);
        const float wx = gx - x0f, wy = gy - y0f;
        const int x0 = (int)x0f, y0 = (int)y0f;

        // clamped reads land on the zero border whenever out of range,
        // which exactly reproduces the reference's validity mask.
        const int xc0 = min(max(x0, 0), WI - 1);
        const int xc1 = min(max(x0 + 1, 0), WI - 1);
        const int yc0 = min(max(y0, 0), HI - 1);
        const int yc1 = min(max(y0 + 1, 0), HI - 1);

        const size_t rb = (size_t)n * HI * WI * CC + cbase;
        const float4 v00 = *(const float4*)(img_t + rb + (size_t)(yc0 * WI + xc0) * CC);
        const float4 v10 = *(const float4*)(img_t + rb + (size_t)(yc0 * WI + xc1) * CC);
        const float4 v01 = *(const float4*)(img_t + rb + (size_t)(yc1 * WI + xc0) * CC);
        const float4 v11 = *(const float4*)(img_t + rb + (size_t)(yc1 * WI + xc1) * CC);

        const float w00 = (1.f - wx) * (1.f - wy) * m;
        const float w10 = wx * (1.f - wy) * m;
        const float w01 = (1.f - wx) * wy * m;
        const float w11 = wx * wy * m;

        ax = fmaf(w00, v00.x, fmaf(w10, v10.x, fmaf(w01, v01.x, fmaf(w11, v11.x, ax))));
        ay = fmaf(w00, v00.y, fmaf(w10, v10.y, fmaf(w01, v01.y, fmaf(w11, v11.y, ay))));
        az = fmaf(w00, v00.z, fmaf(w10, v10.z, fmaf(w01, v01.z, fmaf(w11, v11.z, az))));
        aw = fmaf(w00, v00.w, fmaf(w10, v10.w, fmaf(w01, v01.w, fmaf(w11, v11.w, aw))));
    }

    const size_t ob = (size_t)(n * CC + cbase) * HWSZ + pixoff;
    out[ob]            = ax;
    out[ob + HWSZ]     = ay;
    out[ob + 2 * HWSZ] = az;
    out[ob + 3 * HWSZ] = aw;
}

// ---------------------------------------------------------------------------
// Fallback (no workspace): direct NCHW gather, scalar per channel.
// ---------------------------------------------------------------------------
__global__ __launch_bounds__(256) void deform_fallback(const float* __restrict__ img1,
                                                       const float* __restrict__ kv,
                                                       const float* __restrict__ kh,
                                                       const float* __restrict__ off,
                                                       const float* __restrict__ wgt,
                                                       float* __restrict__ out) {
    const int b = blockIdx.x;
    const int pixblock = b & 4095;          // HW/16
    const int g = (b >> 12) & 3;
    const int n = b >> 14;
    const int t = threadIdx.x;
    const int cc = t & 15;
    const int pl = t >> 4;
    const int lin = pixblock * 16 + pl;
    const int wo = lin & 255;
    const int ho = lin >> 8;
    const int pixoff = lin;
    const int c = g * GC + cc;

    float kvv[3], khh[3];
#pragma unroll
    for (int j = 0; j < 3; ++j) kvv[j] = kv[(n * 3 + j) * HWSZ + pixoff];
#pragma unroll
    for (int i = 0; i < 3; ++i) khh[i] = kh[(n * 12 + g * 3 + i) * HWSZ + pixoff];

    const float* imc = img1 + (size_t)(n * CC + c) * HWSZ;
    float acc = 0.f;
#pragma unroll
    for (int p = 0; p < 9; ++p) {
        const int i = p / 3, j = p % 3;
        const int q = g * 9 + p;
        const float ox = off[(n * 72 + 2 * q)     * HWSZ + pixoff];
        const float oy = off[(n * 72 + 2 * q + 1) * HWSZ + pixoff];
        const float m  = wgt[(n * 36 + q) * HWSZ + pixoff] * khh[i] * kvv[j];

        const float gx = (float)(wo + i) + ox;
        const float gy = (float)(ho + j) + oy;
        const float x0f = floorf(gx), y0f = floorf(gy);
        const float wx = gx - x0f, wy = gy - y0f;
        const int x0 = (int)x0f, y0 = (int)y0f;

        float v[4];
#pragma unroll
        for (int k = 0; k < 4; ++k) {
            const int xs = x0 + (k & 1) - 1;   // padded->original coords
            const int ys = y0 + (k >> 1) - 1;
            v[k] = ((unsigned)xs < (unsigned)WW && (unsigned)ys < (unsigned)HH)
                       ? imc[ys * WW + xs] : 0.f;
        }
        const float sv = (1.f - wx) * (1.f - wy) * v[0] + wx * (1.f - wy) * v[1]
                       + (1.f - wx) * wy * v[2]         + wx * wy * v[3];
        acc = fmaf(m, sv, acc);
    }
    out[(size_t)(n * CC + c) * HWSZ + pixoff] = acc;
}

extern "C" void kernel_launch(void* const* d_in, const int* in_sizes, int n_in,
                              void* d_out, int out_size, void* d_ws, size_t ws_size,
                              hipStream_t stream) {
    const float* img1 = (const float*)d_in[0];
    const float* kv   = (const float*)d_in[1];
    const float* kh   = (const float*)d_in[2];
    const float* off  = (const float*)d_in[3];
    const float* wgt  = (const float*)d_in[4];
    float* out = (float*)d_out;

    const size_t need = (size_t)NN * HI * WI * CC * sizeof(float);  // ~34 MB
    if (ws_size >= need) {
        float* img_t = (float*)d_ws;
        const int n4 = (NN * HI * WI * CC) / 4;
        zero_f4<<<2048, 256, 0, stream>>>((float4*)img_t, n4);
        pad_transpose<<<NN * HH * (WW / 64), 256, 0, stream>>>(img1, img_t);
        deform_main<<<NN * GG * (HWSZ / 64), 256, 0, stream>>>(img_t, kv, kh, off, wgt, out);
    } else {
        deform_fallback<<<NN * GG * (HWSZ / 16), 256, 0, stream>>>(img1, kv, kh, off, wgt, out);
    }
}